// Extract_Overall_4672924418326
// MI455X (gfx1250) — compile-verified
//
#include <hip/hip_runtime.h>
#include <hip/hip_bf16.h>

typedef __attribute__((ext_vector_type(2))) float v2f;
typedef __attribute__((ext_vector_type(8))) float v8f;

#define HD 128            // hidden dim D
#define LDS_STRIDE 132    // 128 + 4 pad: 528B row stride, 16B aligned, conflict-free

// -------------------------------------------------------------------------
// Kernel 1: h = feature @ W^T + bias  via V_WMMA_F32_16X16X4_F32 (exact fp32)
// Block = 128 threads (4 waves). Block owns a 16-row M-tile; wave w owns
// column tiles [32w, 32w+32). K=128 -> 32 WMMA steps of K=4 per col tile.
// -------------------------------------------------------------------------
__global__ void gemm_wmma_f32(const float* __restrict__ feat,
                              const float* __restrict__ W,
                              const float* __restrict__ bias,
                              float* __restrict__ h, int nrows) {
    __shared__ float As[16][LDS_STRIDE];   // 16 x 128 A-tile (padded)

    const int t    = threadIdx.x;          // 0..127
    const int wave = t >> 5;               // 0..3
    const int lane = t & 31;
    const int rowBase = blockIdx.x * 16;

    // Cooperative A-tile load: 16*128 floats = 512 float4, 4 per thread.
    #pragma unroll
    for (int i = 0; i < 4; ++i) {
        int idx = t + i * 128;             // float4 index
        int row = idx >> 5;                // 32 float4 per row
        int c4  = idx & 31;
        float4 v = make_float4(0.f, 0.f, 0.f, 0.f);
        if (rowBase + row < nrows)
            v = *(const float4*)(feat + (long long)(rowBase + row) * HD + c4 * 4);
        *(float4*)(&As[row][c4 * 4]) = v;
    }
    __syncthreads();

    const int m  = lane & 15;              // A row within tile / B col N
    const int hf = lane >> 4;              // lane-half selects K sub-pair
    const int n  = m;
    const int col0 = wave * 32;
    const int col1 = col0 + 16;

    v8f acc0 = {}; v8f acc1 = {};

    #pragma unroll 4
    for (int k0 = 0; k0 < HD; k0 += 4) {
        // A fragment (16x4 f32): VGPR0 = K{0|2}, VGPR1 = K{1|3} per lane-half
        v2f a;
        a.x = As[m][k0 + 2 * hf];
        a.y = As[m][k0 + 2 * hf + 1];
        // B fragment (4x16 f32): B(k, nc) = W[(colBase+nc)*128 + k]
        const float* wc0 = W + (long long)(col0 + n) * HD + k0 + 2 * hf;
        const float* wc1 = W + (long long)(col1 + n) * HD + k0 + 2 * hf;
        v2f b0, b1;
        b0.x = wc0[0]; b0.y = wc0[1];
        b1.x = wc1[0]; b1.y = wc1[1];
        acc0 = __builtin_amdgcn_wmma_f32_16x16x4_f32(false, a, false, b0,
                                                     (short)0, acc0, false, false);
        acc1 = __builtin_amdgcn_wmma_f32_16x16x4_f32(false, a, false, b1,
                                                     (short)0, acc1, false, false);
    }

    // Epilogue: C layout -> VGPR r holds M = r + 8*hf, N = lane&15. Add bias.
    const float bz0 = bias[col0 + n];
    const float bz1 = bias[col1 + n];
    #pragma unroll
    for (int r = 0; r < 8; ++r) {
        int M = r + 8 * hf;
        if (rowBase + M < nrows) {
            h[(long long)(rowBase + M) * HD + col0 + n] = acc0[r] + bz0;
            h[(long long)(rowBase + M) * HD + col1 + n] = acc1[r] + bz1;
        }
    }
}

// -------------------------------------------------------------------------
// Kernel 2: edge scatter  agg[dst] += val * h[src]
// 32 lanes per edge, float4 gather (global_load_b128) + 4 f32 atomics.
// h and agg are both L2-resident (51 MB each vs 192 MB L2).
// -------------------------------------------------------------------------
__global__ void edge_scatter(const float* __restrict__ h,
                             const int* __restrict__ esrc,
                             const int* __restrict__ edst,
                             const float* __restrict__ eval,
                             float* __restrict__ agg, int E) {
    long long gid = (long long)blockIdx.x * blockDim.x + threadIdx.x;
    int e = (int)(gid >> 5);
    if (e >= E) return;
    int lane = (int)(gid & 31);

    int src = esrc[e];
    int dst = edst[e];
    float v = eval[e];

    float4 hv = *(const float4*)(h + (long long)src * HD + lane * 4);
    float* a = agg + (long long)dst * HD + lane * 4;
    atomicAdd(a + 0, v * hv.x);
    atomicAdd(a + 1, v * hv.y);
    atomicAdd(a + 2, v * hv.z);
    atomicAdd(a + 3, v * hv.w);
}

// -------------------------------------------------------------------------
// Kernel 3: out[d] = mean_n relu(agg[n][d])   (blockDim = 128 = D)
// -------------------------------------------------------------------------
__global__ void relu_mean(const float* __restrict__ agg,
                          float* __restrict__ out, int nrows) {
    int d = threadIdx.x;
    float acc = 0.f;
    for (int r = blockIdx.x; r < nrows; r += gridDim.x)
        acc += fmaxf(agg[(long long)r * HD + d], 0.f);
    atomicAdd(&out[d], acc * (1.0f / (float)nrows));
}

__global__ void zero_f32(float* __restrict__ p, long long n) {
    long long i = (long long)blockIdx.x * blockDim.x + threadIdx.x;
    long long stride = (long long)gridDim.x * blockDim.x;
    for (; i < n; i += stride) p[i] = 0.f;
}

extern "C" void kernel_launch(void* const* d_in, const int* in_sizes, int n_in,
                              void* d_out, int out_size, void* d_ws, size_t ws_size,
                              hipStream_t stream) {
    const float* feat = (const float*)d_in[0];
    const int*   esrc = (const int*)d_in[1];
    const int*   edst = (const int*)d_in[2];
    const float* eval = (const float*)d_in[3];
    const float* W    = (const float*)d_in[4];
    const float* bias = (const float*)d_in[5];

    const int N = in_sizes[0] / HD;
    const int E = in_sizes[1];

    if (ws_size < (size_t)2 * (size_t)N * HD * sizeof(float)) return;
    float* h   = (float*)d_ws;                 // [N, 128]
    float* agg = h + (size_t)N * HD;           // [N, 128]
    float* out = (float*)d_out;                // [128]

    // Zero accumulators (harness poisons d_out/d_ws; no re-poison per replay).
    zero_f32<<<2048, 256, 0, stream>>>(agg, (long long)N * HD);
    zero_f32<<<1, 128, 0, stream>>>(out, HD);

    // GEMM: one block per 16-row tile.
    int mtiles = (N + 15) / 16;
    gemm_wmma_f32<<<mtiles, 128, 0, stream>>>(feat, W, bias, h, N);

    // Scatter: 32 lanes per edge.
    long long sthreads = (long long)E * 32;
    int sblocks = (int)((sthreads + 255) / 256);
    edge_scatter<<<sblocks, 256, 0, stream>>>(h, esrc, edst, eval, agg, E);

    // ReLU + column mean.
    relu_mean<<<512, HD, 0, stream>>>(agg, out, N);
}